// TDiscriminator_56968446214147
// MI455X (gfx1250) — compile-verified
//
#include <hip/hip_runtime.h>

typedef __attribute__((ext_vector_type(2))) float v2f;
typedef __attribute__((ext_vector_type(8))) float v8f;

#define HID    128
#define T_LEN  32768
#define BATCH  4
#define LC     128
#define HOP    256
#define KER    3
#define NL     4
#define CONDC  512
#define CKTOT  (HID * KER)      // 384
#define MROWS  (HID * CKTOT)    // 49152 rows per layer (o*384 + c*3 + k)
#define SLOPE_PRE 0.2f
#define SLOPE_BLK 3.0f          // reference passes ker_size(3) as the slope

__device__ __forceinline__ v8f wmma4(v2f a, v2f b, v8f c) {
  // D = A(16x4 f32) * B(4x16 f32) + C(16x16 f32)
  return __builtin_amdgcn_wmma_f32_16x16x4_f32(false, a, false, b, (short)0, c,
                                               false, false);
}
__device__ __forceinline__ float leaky(float x, float s) {
  return x >= 0.0f ? x : s * x;
}

// ---------------------------------------------------------------- pre conv
// h[b][ch][t] = leaky(x[b][t]*pw[ch] + pb[ch], 0.2)
__global__ __launch_bounds__(256) void pre_kernel(const float* __restrict__ x,
                                                  const float* __restrict__ pw,
                                                  const float* __restrict__ pb,
                                                  float* __restrict__ h) {
  int idx = blockIdx.x * 256 + threadIdx.x;    // (b*128 + ch)*T + t
  int t  = idx & (T_LEN - 1);
  int bc = idx >> 15;
  int ch = bc & (HID - 1);
  int b  = bc >> 7;
  float v = x[(size_t)b * T_LEN + t] * pw[ch] + pb[ch];
  h[idx] = leaky(v, SLOPE_PRE);
}

// ---------------------------------------------------------------- bias predictor
// bws[b][m][l] = bb[m] + sum_q bw[m][q] * c[b][q][l]   (m = layer*128 + o)
__global__ __launch_bounds__(256) void biaspred_kernel(const float* __restrict__ bw,
                                                       const float* __restrict__ bb,
                                                       const float* __restrict__ c,
                                                       float* __restrict__ bws) {
  int idx = blockIdx.x * 256 + threadIdx.x;    // (b*512 + m)*128 + l
  int l = idx & (LC - 1);
  int m = (idx >> 7) & (CONDC - 1);
  int b = idx >> 16;
  const float* wrow = bw + (size_t)m * CONDC;
  const float* ccol = c + (size_t)b * CONDC * LC + l;
  float acc = bb[m];
  for (int q = 0; q < CONDC; ++q) acc += wrow[q] * ccol[(size_t)q * LC];
  bws[idx] = acc;
}

// ---------------------------------------------------------------- kernel predictor GEMM
// Per (b,layer): kws[b][layer][l][m] = kb[r(m)] + sum_q kw[r(m)][q]*c[b][q][l]
// with m = o*384 + ci*3 + k  and  r = ((layer*128+ci)*128+o)*3 + k
// (row permutation folded into A addressing; output contiguous in LVC W-layout)
__global__ __launch_bounds__(256) void kpred_gemm(const float* __restrict__ kw,
                                                  const float* __restrict__ kb,
                                                  const float* __restrict__ c,
                                                  float* __restrict__ kws) {
  __shared__ __align__(16) float As[128][36];  // A chunk: 128 m-rows x 32 K
  __shared__ __align__(16) float Bs[128][36];  // B chunk transposed: 128 l x 32 K
  const int mb = blockIdx.x, layer = blockIdx.y, b = blockIdx.z;
  const int tid = threadIdx.x;
  const int wave = tid >> 5, lane = tid & 31;
  const int m0 = mb * 128;

  // A staging map: thread -> (row in chunk, 16-col segment)
  const int arow  = tid >> 1;
  const int acseg = (tid & 1) * 16;
  {
    int m  = m0 + arow;
    int o  = m / CKTOT;
    int rm = m - o * CKTOT;
    int ci = rm / KER;
    int kk = rm - ci * KER;
    long r = (((long)layer * HID + ci) * HID + o) * KER + kk;
    // cache per-thread global row pointer in a register
    ((void)r);
  }
  int mA  = m0 + arow;
  int oA  = mA / CKTOT;
  int rmA = mA - oA * CKTOT;
  int ciA = rmA / KER;
  int kkA = rmA - ciA * KER;
  const float* aptr = kw + ((((long)layer * HID + ciA) * HID + oA) * KER + kkA) * CONDC + acseg;

  // B staging map: thread -> (K row of chunk, 16-l segment)
  const int bqq   = tid >> 3;
  const int blseg = (tid & 7) * 16;
  const float* cb = c + (size_t)b * CONDC * LC;

  v8f acc[8];
  #pragma unroll
  for (int i = 0; i < 8; ++i)
    #pragma unroll
    for (int j = 0; j < 8; ++j) acc[i][j] = 0.0f;

  for (int chk = 0; chk < CONDC / 32; ++chk) {
    const int q0 = chk * 32;
    // stage A (coalesced 16B loads along K)
    {
      const float4* src = (const float4*)(aptr + q0);
      float4* dst = (float4*)&As[arow][acseg];
      #pragma unroll
      for (int j = 0; j < 4; ++j) dst[j] = src[j];
    }
    // stage B transposed: Bs[l][qq] = c[b][q0+qq][l]
    {
      const float4* src = (const float4*)(cb + (size_t)(q0 + bqq) * LC + blseg);
      float4 f0 = src[0], f1 = src[1], f2 = src[2], f3 = src[3];
      float tmp[16] = {f0.x,f0.y,f0.z,f0.w, f1.x,f1.y,f1.z,f1.w,
                       f2.x,f2.y,f2.z,f2.w, f3.x,f3.y,f3.z,f3.w};
      #pragma unroll
      for (int j = 0; j < 16; ++j) Bs[blseg + j][bqq] = tmp[j];
    }
    __syncthreads();
    #pragma unroll
    for (int qs = 0; qs < 8; ++qs) {
      const int kcol = qs * 4 + ((lane >> 4) << 1);
      v2f a = *(const v2f*)&As[(wave << 4) + (lane & 15)][kcol];
      #pragma unroll
      for (int nt = 0; nt < 8; ++nt) {
        v2f bv = *(const v2f*)&Bs[(nt << 4) + (lane & 15)][kcol];
        acc[nt] = wmma4(a, bv, acc[nt]);
      }
    }
    __syncthreads();
  }

  // epilogue: + kernel bias, store 2x float4 per n-tile (32B contiguous per lane)
  const int hi = lane >> 4;
  const int mbase = m0 + (wave << 4) + (hi << 3);
  float biasv[8];
  #pragma unroll
  for (int v = 0; v < 8; ++v) {
    int mm = mbase + v;
    int oo = mm / CKTOT;
    int rr = mm - oo * CKTOT;
    int cc = rr / KER;
    int k2 = rr - cc * KER;
    long r = (((long)layer * HID + cc) * HID + oo) * KER + k2;
    biasv[v] = kb[r];
  }
  const long blbase = (long)(b * NL + layer) * LC;
  #pragma unroll
  for (int nt = 0; nt < 8; ++nt) {
    int l = (nt << 4) + (lane & 15);
    float* dst = kws + (blbase + l) * (long)MROWS + mbase;
    float4 lo = {acc[nt][0] + biasv[0], acc[nt][1] + biasv[1],
                 acc[nt][2] + biasv[2], acc[nt][3] + biasv[3]};
    float4 hi4 = {acc[nt][4] + biasv[4], acc[nt][5] + biasv[5],
                  acc[nt][6] + biasv[6], acc[nt][7] + biasv[7]};
    *(float4*)(dst) = lo;
    *(float4*)(dst + 4) = hi4;
  }
}

// ---------------------------------------------------------------- LVC / dilated conv
// out[b][o][t0+s] = leaky( bias[o(,l)] + sum_{c,k} W[o][c*3+k]*h[b][c][t0+s+(k-1)*dil], 3.0 )
// LVC: W per (b,l) block in kws; conv: shared block_w (strides 0), dilation d.
__global__ __launch_bounds__(256) void lvc_conv_kernel(
    const float* __restrict__ hin, const float* __restrict__ Wbase,
    long w_b_stride, long w_l_stride, const float* __restrict__ bias,
    long bias_b_stride, int bias_o_stride, int bias_l_stride, int dil,
    float* __restrict__ hout) {
  __shared__ __align__(16) float As[128][36];  // weights: 128 o x 32 ck
  __shared__ __align__(16) float Bs[256][36];  // windows transposed: 256 s x 32 ck
  const int l = blockIdx.x;    // hop block / time block (256 samples)
  const int b = blockIdx.z;
  const int tid = threadIdx.x;
  const int wave = tid >> 5, lane = tid & 31;
  const int t0 = l * HOP;

  const float* W  = Wbase + (long)b * w_b_stride + (long)l * w_l_stride;
  const float* hb = hin + (size_t)b * HID * T_LEN;

  v8f acc[16];
  #pragma unroll
  for (int i = 0; i < 16; ++i)
    #pragma unroll
    for (int j = 0; j < 8; ++j) acc[i][j] = 0.0f;

  const int arow  = tid >> 1;
  const int acseg = (tid & 1) * 16;

  for (int chk = 0; chk < CKTOT / 32; ++chk) {   // 12 chunks of K=32
    const int ck0 = chk * 32;
    // stage weights (coalesced 16B along ck)
    {
      const float4* src = (const float4*)(W + (long)arow * CKTOT + ck0 + acseg);
      float4* dst = (float4*)&As[arow][acseg];
      #pragma unroll
      for (int j = 0; j < 4; ++j) dst[j] = src[j];
    }
    // stage input windows transposed, zero-padded at sequence edges
    #pragma unroll 4
    for (int it = 0; it < 32; ++it) {
      int e   = it * 256 + tid;     // covers 32 ck-rows x 256 s, s fastest
      int qq  = e >> 8;
      int s   = e & 255;
      int ckg = ck0 + qq;
      int cch = ckg / KER;
      int tap = ckg - cch * KER;
      int tt  = t0 + s + (tap - 1) * dil;
      float v = 0.0f;
      if ((unsigned)tt < (unsigned)T_LEN) v = hb[(size_t)cch * T_LEN + tt];
      Bs[s][qq] = v;
    }
    __syncthreads();
    #pragma unroll
    for (int qs = 0; qs < 8; ++qs) {
      const int kcol = qs * 4 + ((lane >> 4) << 1);
      v2f a = *(const v2f*)&As[(wave << 4) + (lane & 15)][kcol];
      #pragma unroll
      for (int nt = 0; nt < 16; ++nt) {
        v2f bv = *(const v2f*)&Bs[(nt << 4) + (lane & 15)][kcol];
        acc[nt] = wmma4(a, bv, acc[nt]);
      }
    }
    __syncthreads();
  }

  // epilogue: bias + leaky(3.0), store (lanes 0-15 contiguous in t per o-row)
  const int hi = lane >> 4;
  const int obase = (wave << 4) + (hi << 3);
  float biasv[8];
  #pragma unroll
  for (int v = 0; v < 8; ++v)
    biasv[v] = bias[(long)b * bias_b_stride + (long)(obase + v) * bias_o_stride +
                    (long)l * bias_l_stride];
  #pragma unroll
  for (int nt = 0; nt < 16; ++nt) {
    int s = (nt << 4) + (lane & 15);
    long t = t0 + s;
    #pragma unroll
    for (int v = 0; v < 8; ++v) {
      float y = acc[nt][v] + biasv[v];
      hout[(size_t)(b * HID + obase + v) * T_LEN + t] = leaky(y, SLOPE_BLK);
    }
  }
}

// ---------------------------------------------------------------- post conv
__global__ __launch_bounds__(256) void post_kernel(const float* __restrict__ h,
                                                   const float* __restrict__ pw,
                                                   const float* __restrict__ pb,
                                                   float* __restrict__ out) {
  int idx = blockIdx.x * 256 + threadIdx.x;    // b*T + t
  int t = idx & (T_LEN - 1);
  int b = idx >> 15;
  const float* hc = h + (size_t)b * HID * T_LEN + t;
  float acc = pb[0];
  for (int o = 0; o < HID; ++o) acc += pw[o] * hc[(size_t)o * T_LEN];
  out[idx] = acc;
}

// ---------------------------------------------------------------- launch
extern "C" void kernel_launch(void* const* d_in, const int* in_sizes, int n_in,
                              void* d_out, int out_size, void* d_ws, size_t ws_size,
                              hipStream_t stream) {
  (void)in_sizes; (void)n_in; (void)out_size; (void)ws_size;
  const float* x   = (const float*)d_in[0];
  const float* c   = (const float*)d_in[1];
  const float* kw  = (const float*)d_in[2];
  const float* kb  = (const float*)d_in[3];
  const float* bw  = (const float*)d_in[4];
  const float* bb  = (const float*)d_in[5];
  const float* prw = (const float*)d_in[6];
  const float* prb = (const float*)d_in[7];
  const float* blw = (const float*)d_in[8];
  const float* blb = (const float*)d_in[9];
  const float* pow_ = (const float*)d_in[10];
  const float* pob  = (const float*)d_in[11];
  float* out = (float*)d_out;

  float* kws = (float*)d_ws;                              // [B][NL][LC][MROWS]
  float* bws = kws + (size_t)BATCH * NL * LC * MROWS;     // [B][512][LC]
  float* ha  = bws + (size_t)BATCH * NL * HID * LC;       // [B][HID][T]
  float* hb2 = ha + (size_t)BATCH * HID * T_LEN;          // [B][HID][T]

  pre_kernel<<<(BATCH * HID * T_LEN) / 256, 256, 0, stream>>>(x, prw, prb, ha);
  biaspred_kernel<<<(BATCH * CONDC * LC) / 256, 256, 0, stream>>>(bw, bb, c, bws);
  kpred_gemm<<<dim3(MROWS / 128, NL, BATCH), 256, 0, stream>>>(kw, kb, c, kws);

  float* cur = ha;
  float* nxt = hb2;
  const int dils[NL] = {1, 3, 9, 27};
  for (int i = 0; i < NL; ++i) {
    // LVC block (per-location kernels, dilation 1)
    lvc_conv_kernel<<<dim3(LC, 1, BATCH), 256, 0, stream>>>(
        cur, kws + (long)i * LC * MROWS, (long)NL * LC * MROWS, (long)MROWS,
        bws + (long)i * HID * LC, (long)CONDC * LC, LC, 1, 1, nxt);
    float* t0 = cur; cur = nxt; nxt = t0;
    // dilated dense conv (shared weights)
    lvc_conv_kernel<<<dim3(T_LEN / HOP, 1, BATCH), 256, 0, stream>>>(
        cur, blw + (long)i * MROWS, 0L, 0L,
        blb + (long)i * HID, 0L, 1, 0, dils[i], nxt);
    float* t1 = cur; cur = nxt; nxt = t1;
  }
  post_kernel<<<(BATCH * T_LEN) / 256, 256, 0, stream>>>(cur, pow_, pob, out);
}